// TreeLSTM_9431748182481
// MI455X (gfx1250) — compile-verified
//
#include <hip/hip_runtime.h>
#include <math.h>

// ---------------------------------------------------------------------------
// TreeLSTM on MI455X (gfx1250, wave32). fp32 WMMA 16x16x4 path.
// Memory-bound (~1GB traffic vs ~70 GFLOP): keep fp32 precision, ride the
// matrix pipes with v_wmma_f32_16x16x4_f32, and fuse the mean-pooling partial
// sums into the compute kernels so h_buf is never re-read (saves ~268MB).
// ---------------------------------------------------------------------------

typedef __attribute__((ext_vector_type(2))) float v2f;
typedef __attribute__((ext_vector_type(8))) float v8f;

#define H        128
#define H3       384
#define DEPTH    14
#define NPT      32767        // nodes per tree
#define NTREE    16
#define LEAF0    16383        // first leaf local id = 2^14 - 1
#define NLEAF_TREE 16384
#define NSLOT    2051         // pooling partial slots per tree: 1024 + 1023 + 4

__device__ __forceinline__ float sigm(float x) { return 1.0f / (1.0f + __expf(-x)); }

__device__ __forceinline__ v8f wmma4(v2f a, v2f b, v8f c) {
  // D = A(16x4,f32) * B(4x16,f32) + C(16x16,f32)
  return __builtin_amdgcn_wmma_f32_16x16x4_f32(
      /*neg_a=*/false, a, /*neg_b=*/false, b,
      /*c_mod=*/(short)0, c, /*reuse_a=*/false, /*reuse_b=*/false);
}

// ---------------------------------------------------------------------------
// Leaf pass: iou = feature @ W_iou^T + b ; gates with c_in = c_init.
// One block = 16 leaves (contiguous within a tree). 8 waves, 3 col-tiles each.
// Fused: per-block column partial sum of h_new -> pooling slot.
// ---------------------------------------------------------------------------
__global__ __launch_bounds__(256) void leaf_kernel(
    const float* __restrict__ feat, const float* __restrict__ c_init,
    const float* __restrict__ W_iou, const float* __restrict__ b_iou,
    float* __restrict__ h_buf, float* __restrict__ c_buf,
    float* __restrict__ ppart)
{
  __shared__ __align__(16) float sA[16][H];      // 8 KB feature panel
  __shared__ __align__(16) float sIOU[16][H3];   // 24 KB iou panel
  __shared__ float sPart[256];

  const int tid = threadIdx.x;
  const int gi = blockIdx.x * 16;                // flat leaf index
  const int tree = gi >> 14;
  const int li = gi & (NLEAF_TREE - 1);
  const int slot = blockIdx.x & 1023;            // 1024 leaf blocks per tree
  const long nodebase = (long)tree * NPT + LEAF0 + li;

  // stage 16x128 feature panel (float4, fully coalesced)
  for (int i = 0; i < 2; ++i) {
    int off = (tid + i * 256) * 4;
    int row = off >> 7, col = off & 127;
    *(float4*)&sA[row][col] = *(const float4*)(feat + (nodebase + row) * H + col);
  }
  __syncthreads();

  const int wave = tid >> 5, lane = tid & 31;
  const int m = lane & 15, kp = lane >> 4;       // A: row / k-pair ; B: col / k-pair
  const int c0 = (wave * 3 + 0) * 16;
  const int c1 = (wave * 3 + 1) * 16;
  const int c2 = (wave * 3 + 2) * 16;

  v8f a0 = {}, a1 = {}, a2 = {};
  #pragma unroll 4
  for (int k0 = 0; k0 < H; k0 += 4) {
    v2f af = *(const v2f*)&sA[m][k0 + 2 * kp];
    v2f b0 = *(const v2f*)(W_iou + (c0 + m) * H + k0 + 2 * kp);
    v2f b1 = *(const v2f*)(W_iou + (c1 + m) * H + k0 + 2 * kp);
    v2f b2 = *(const v2f*)(W_iou + (c2 + m) * H + k0 + 2 * kp);
    a0 = wmma4(af, b0, a0);
    a1 = wmma4(af, b1, a1);
    a2 = wmma4(af, b2, a2);
  }
  #pragma unroll
  for (int v = 0; v < 8; ++v) {
    int row = v + 8 * kp;                        // C layout: M = v + 8*(lane>=16)
    sIOU[row][c0 + m] = a0[v] + b_iou[c0 + m];
    sIOU[row][c1 + m] = a1[v] + b_iou[c1 + m];
    sIOU[row][c2 + m] = a2[v] + b_iou[c2 + m];
  }
  __syncthreads();

  float hsum = 0.f;                              // col = tid&127 for all 8 iters
  for (int i = 0; i < 8; ++i) {
    int idx = tid + i * 256;
    int row = idx >> 7, col = idx & 127;
    float ig = sIOU[row][col];
    float og = sIOU[row][col + H];
    float ug = sIOU[row][col + 2 * H];
    float cin = c_init[(nodebase + row) * H + col];
    float cN = sigm(ig) * tanhf(ug) + cin;
    float hN = sigm(og) * tanhf(cN);
    h_buf[(nodebase + row) * H + col] = hN;
    c_buf[(nodebase + row) * H + col] = cN;
    hsum += hN;
  }
  sPart[tid] = hsum;
  __syncthreads();
  if (tid < 128)
    ppart[((long)tree * NSLOT + slot) * H + tid] = sPart[tid] + sPart[tid + 128];
}

// ---------------------------------------------------------------------------
// Internal level pass (fused): children gather, forget-gate GEMM, child-pair
// reduction, iou GEMM, gates, pooling partial. One block = up to 16 nodes of
// one tree; the 32 children are the contiguous id range [2*local0+1, ...).
// ---------------------------------------------------------------------------
__global__ __launch_bounds__(256) void level_kernel(
    const float* __restrict__ U_iou, const float* __restrict__ b_iou,
    const float* __restrict__ U_f_w, const float* __restrict__ U_f_b,
    float* __restrict__ h_buf, float* __restrict__ c_buf,
    float* __restrict__ ppart, int l, int slotbase)
{
  __shared__ __align__(16) float sCh[32][H];     // 16 KB children h
  __shared__ __align__(16) float sHt[16][H];     //  8 KB child-sum h~
  __shared__ __align__(16) float sCagg[16][H];   //  8 KB sum_k f_k * c_k
  __shared__ __align__(16) float sIOU[16][H3];   // 24 KB iou
  __shared__ float sPart[256];

  const int tid = threadIdx.x;
  const int levN = 1 << l;
  const int bpt = (levN + 15) >> 4;              // blocks per tree
  const int tree = blockIdx.x / bpt;
  const int jb = blockIdx.x - tree * bpt;
  const int local0 = levN - 1 + 16 * jb;
  const int nvalid = min(16, levN - 16 * jb);    // <16 only for levels 0..3
  const int nchild = 2 * nvalid;
  const long nodebase  = (long)tree * NPT + local0;
  const long childbase = (long)tree * NPT + 2 * (long)local0 + 1;

  // stage children h (zero-padded beyond nchild)
  for (int i = 0; i < 4; ++i) {
    int off = (tid + i * 256) * 4;
    int row = off >> 7, col = off & 127;
    float4 val; val.x = val.y = val.z = val.w = 0.f;
    if (row < nchild) val = *(const float4*)(h_buf + (childbase + row) * H + col);
    *(float4*)&sCh[row][col] = val;
  }
  __syncthreads();

  // h_tild = h_left + h_right (consumed by GEMM2, after the next barrier)
  for (int i = 0; i < 8; ++i) {
    int idx = tid + i * 256;
    int row = idx >> 7, col = idx & 127;
    sHt[row][col] = sCh[2 * row][col] + sCh[2 * row + 1][col];
  }

  const int wave = tid >> 5, lane = tid & 31;
  const int m = lane & 15, kp = lane >> 4;

  // ---- GEMM1: f_pre = children_h[32,128] @ U_f^T  (16 tiles, 2 per wave)
  const int rt  = wave >> 2;                     // row-tile 0/1
  const int ct0 = (wave & 3) * 16;               // col tiles ct0 and ct0+64
  v8f f0 = {}, f1 = {};
  #pragma unroll 4
  for (int k0 = 0; k0 < H; k0 += 4) {
    v2f af = *(const v2f*)&sCh[rt * 16 + m][k0 + 2 * kp];
    v2f b0 = *(const v2f*)(U_f_w + (ct0 + m) * H + k0 + 2 * kp);
    v2f b1 = *(const v2f*)(U_f_w + (ct0 + 64 + m) * H + k0 + 2 * kp);
    f0 = wmma4(af, b0, f0);
    f1 = wmma4(af, b1, f1);
  }
  // ---- fuse: c_agg[node] = sigm(f_L)*c_L + sigm(f_R)*c_R.
  // Child-pair rows (2j,2j+1) sit in adjacent accumulator regs of one lane.
  {
    float bias0 = U_f_b[ct0 + m];
    float bias1 = U_f_b[ct0 + 64 + m];
    #pragma unroll
    for (int v = 0; v < 8; v += 2) {
      int rowA = rt * 16 + v + 8 * kp;           // left child row
      int nrow = rt * 8 + (v >> 1) + 4 * kp;     // node row
      float cA = (rowA     < nchild) ? c_buf[(childbase + rowA    ) * H + (ct0 + m)] : 0.f;
      float cB = (rowA + 1 < nchild) ? c_buf[(childbase + rowA + 1) * H + (ct0 + m)] : 0.f;
      sCagg[nrow][ct0 + m] = sigm(f0[v] + bias0) * cA + sigm(f0[v + 1] + bias0) * cB;
      float cA1 = (rowA     < nchild) ? c_buf[(childbase + rowA    ) * H + (ct0 + 64 + m)] : 0.f;
      float cB1 = (rowA + 1 < nchild) ? c_buf[(childbase + rowA + 1) * H + (ct0 + 64 + m)] : 0.f;
      sCagg[nrow][ct0 + 64 + m] = sigm(f1[v] + bias1) * cA1 + sigm(f1[v + 1] + bias1) * cB1;
    }
  }
  __syncthreads();

  // ---- GEMM2: iou = h_tild[16,128] @ U_iou^T  (24 tiles, 3 per wave)
  const int d0 = (wave * 3 + 0) * 16;
  const int d1 = (wave * 3 + 1) * 16;
  const int d2 = (wave * 3 + 2) * 16;
  v8f i0 = {}, i1 = {}, i2 = {};
  #pragma unroll 4
  for (int k0 = 0; k0 < H; k0 += 4) {
    v2f af = *(const v2f*)&sHt[m][k0 + 2 * kp];
    v2f b0 = *(const v2f*)(U_iou + (d0 + m) * H + k0 + 2 * kp);
    v2f b1 = *(const v2f*)(U_iou + (d1 + m) * H + k0 + 2 * kp);
    v2f b2 = *(const v2f*)(U_iou + (d2 + m) * H + k0 + 2 * kp);
    i0 = wmma4(af, b0, i0);
    i1 = wmma4(af, b1, i1);
    i2 = wmma4(af, b2, i2);
  }
  #pragma unroll
  for (int v = 0; v < 8; ++v) {
    int row = v + 8 * kp;
    sIOU[row][d0 + m] = i0[v] + b_iou[d0 + m];
    sIOU[row][d1 + m] = i1[v] + b_iou[d1 + m];
    sIOU[row][d2 + m] = i2[v] + b_iou[d2 + m];
  }
  __syncthreads();

  // ---- gates + fused pooling partial (col = tid&127 for all 8 iters)
  float hsum = 0.f;
  for (int i = 0; i < 8; ++i) {
    int idx = tid + i * 256;
    int row = idx >> 7, col = idx & 127;
    if (row < nvalid) {
      float ig = sIOU[row][col];
      float og = sIOU[row][col + H];
      float ug = sIOU[row][col + 2 * H];
      float cin = sCagg[row][col];
      float cN = sigm(ig) * tanhf(ug) + cin;
      float hN = sigm(og) * tanhf(cN);
      h_buf[(nodebase + row) * H + col] = hN;
      c_buf[(nodebase + row) * H + col] = cN;
      hsum += hN;
    }
  }
  sPart[tid] = hsum;
  __syncthreads();
  if (tid < 128)
    ppart[((long)tree * NSLOT + slotbase + jb) * H + tid] = sPart[tid] + sPart[tid + 128];
}

// ---------------------------------------------------------------------------
// Deterministic slot reduction (fixed order, no float atomics) -> hg mean.
// ---------------------------------------------------------------------------
__global__ __launch_bounds__(128) void pool_reduce_kernel(
    const float* __restrict__ ppart, float* __restrict__ hg)
{
  const int tree = blockIdx.x, col = threadIdx.x;
  const float* p = ppart + (long)tree * NSLOT * H + col;
  float acc = 0.f;
  for (int s = 0; s < NSLOT; ++s) acc += p[(long)s * H];
  hg[tree * H + col] = acc * (1.0f / 32767.0f);
}

__global__ __launch_bounds__(32) void cls_kernel(const float* __restrict__ hg,
    const float* __restrict__ W_lin, const float* __restrict__ b_lin,
    float* __restrict__ out)
{
  __shared__ float lg[10];
  __shared__ float red[2];
  const int t = blockIdx.x, lane = threadIdx.x;
  if (lane < 10) {
    float acc = b_lin[lane];
    for (int k = 0; k < H; ++k) acc += hg[t * H + k] * W_lin[lane * H + k];
    lg[lane] = acc;
  }
  __syncthreads();
  if (lane == 0) {
    float mx = lg[0];
    for (int j = 1; j < 10; ++j) mx = fmaxf(mx, lg[j]);
    float ssum = 0.f;
    for (int j = 0; j < 10; ++j) ssum += __expf(lg[j] - mx);
    red[0] = mx; red[1] = __logf(ssum);
  }
  __syncthreads();
  if (lane < 10) out[t * 10 + lane] = lg[lane] - red[0] - red[1];
}

// ---------------------------------------------------------------------------
extern "C" void kernel_launch(void* const* d_in, const int* in_sizes, int n_in,
                              void* d_out, int out_size, void* d_ws, size_t ws_size,
                              hipStream_t stream) {
  const float* feature = (const float*)d_in[0];
  const float* c_init  = (const float*)d_in[2];
  const float* W_iou   = (const float*)d_in[3];
  const float* U_iou   = (const float*)d_in[4];
  const float* b_iou   = (const float*)d_in[5];
  const float* U_f_w   = (const float*)d_in[6];
  const float* U_f_b   = (const float*)d_in[7];
  const float* W_lin   = (const float*)d_in[8];
  const float* b_lin   = (const float*)d_in[9];
  (void)in_sizes; (void)n_in; (void)out_size; (void)ws_size;

  const long N = (long)NTREE * NPT;
  // Workspace layout (inputs must not be mutated): our own h/c state + pooling.
  float* h_buf = (float*)d_ws;                   // N*H
  float* c_buf = h_buf + N * H;                  // N*H
  float* hg    = c_buf + N * H;                  // 16*H
  float* ppart = hg + NTREE * H;                 // 16*NSLOT*H (~17MB)

  // Leaves: 262144 / 16 = 16384 blocks (slots 0..1023 per tree)
  leaf_kernel<<<NTREE * NLEAF_TREE / 16, 256, 0, stream>>>(
      feature, c_init, W_iou, b_iou, h_buf, c_buf, ppart);

  // Internal levels bottom-up (stream order carries the level dependency)
  int slotbase = 1024;
  for (int l = DEPTH - 1; l >= 0; --l) {
    int bpt = ((1 << l) + 15) >> 4;
    level_kernel<<<NTREE * bpt, 256, 0, stream>>>(
        U_iou, b_iou, U_f_w, U_f_b, h_buf, c_buf, ppart, l, slotbase);
    slotbase += bpt;                             // 1024 + 1023 + 4 = NSLOT
  }

  pool_reduce_kernel<<<NTREE, 128, 0, stream>>>(ppart, hg);
  cls_kernel<<<NTREE, 32, 0, stream>>>(hg, W_lin, b_lin, (float*)d_out);
}